// SAGE_2405181685958
// MI455X (gfx1250) — compile-verified
//
#include <hip/hip_runtime.h>
#include <hip/hip_bf16.h>

typedef float v2f __attribute__((ext_vector_type(2)));
typedef float v4f __attribute__((ext_vector_type(4)));
typedef float v8f __attribute__((ext_vector_type(8)));

// ---------------------------------------------------------------------------
// D = A(16x4) x B(4x16) + C, exact fp32 via CDNA5 WMMA
// ---------------------------------------------------------------------------
__device__ __forceinline__ v8f wmma4_f32(v2f a, v2f b, v8f c) {
    return __builtin_amdgcn_wmma_f32_16x16x4_f32(
        /*neg_a=*/false, a, /*neg_b=*/false, b,
        /*c_mod=*/(short)0, c, /*reuse_a=*/false, /*reuse_b=*/false);
}

// CDNA5 async memory->LDS copy (no VGPR round-trip, tracked by ASYNCcnt).
// VDST slot = per-lane LDS byte address, VADDR = 64-bit global address.
__device__ __forceinline__ void async_load_b128_to_lds(unsigned lds_off,
                                                       const float* gaddr) {
    asm volatile("global_load_async_to_lds_b128 %0, %1, off"
                 :: "v"(lds_off), "v"((unsigned long long)(uintptr_t)gaddr)
                 : "memory");
}
__device__ __forceinline__ void wait_asynccnt0() {
    asm volatile("s_wait_asynccnt 0x0" ::: "memory");
}

// ---------------------------------------------------------------------------
// Utility kernels
// ---------------------------------------------------------------------------
__global__ void zero_f32(float* __restrict__ p, int n) {
    int i = blockIdx.x * blockDim.x + threadIdx.x;
    if (i < n) p[i] = 0.0f;
}

__global__ void deg_count(const int* __restrict__ dst, float* __restrict__ deg, int E) {
    int i = blockIdx.x * blockDim.x + threadIdx.x;
    if (i < E) atomicAdd(&deg[dst[i]], 1.0f);
}

__global__ void rdeg_inv(float* __restrict__ deg, int n) {
    int i = blockIdx.x * blockDim.x + threadIdx.x;
    if (i < n) deg[i] = 1.0f / fmaxf(deg[i], 1.0f);
}

// agg[node][c] *= rdeg[node];  C = 1<<logC
__global__ void scale_rows(float* __restrict__ agg, const float* __restrict__ rdeg,
                           int count, int logC) {
    int i = blockIdx.x * blockDim.x + threadIdx.x;
    if (i < count) agg[i] *= rdeg[i >> logC];
}

// One wave per edge: agg[dst] += feat[src] over C channels (C % 128 == 0).
// float4 row loads (coalesced b128), then 4 f32 atomics per lane.
__global__ void scatter_add(const float* __restrict__ feat, const int* __restrict__ src,
                            const int* __restrict__ dst, float* __restrict__ agg,
                            int E, int C) {
    int wid  = (int)((blockIdx.x * blockDim.x + threadIdx.x) >> 5);
    int lane = threadIdx.x & 31;
    if (wid >= E) return;
    int s = src[wid], d = dst[wid];
    const float* fs = feat + (size_t)s * C;
    float*       ad = agg  + (size_t)d * C;
    for (int c = lane * 4; c < C; c += 128) {
        v4f v = *(const v4f*)(fs + c);
        atomicAdd(ad + c + 0, v.x);
        atomicAdd(ad + c + 1, v.y);
        atomicAdd(ad + c + 2, v.z);
        atomicAdd(ad + c + 3, v.w);
    }
}

// ---------------------------------------------------------------------------
// Fused SAGE linear: out = act( A1 @ Wl^T + A2 @ Wr^T + bias )
//   A1, A2 : [M, K] row-major   Wl, Wr : [N, K] row-major   out : [M, N]
//
// Block = 4 waves = 64x64 output super-tile.
//   - async global->LDS staging per 32-wide K chunk (GLOBAL_LOAD_ASYNC_TO_LDS_B128)
//   - each wave: one 16-col tile x four 16-row tiles, fragments via ds_load_b64
//   - LDS row stride 36 dwords -> conflict-free wave32 b64 reads
// Requires M % 16 == 0, K % 32 == 0, N % 64 == 0.
// ---------------------------------------------------------------------------
#define KC 32
#define LDS_STRIDE 36                    // 32 + 4 pad (dwords)
#define MAT_LDS (64 * LDS_STRIDE)        // dwords per staged matrix

__global__ void sage_gemm_fused(const float* __restrict__ A1, const float* __restrict__ A2,
                                const float* __restrict__ Wl, const float* __restrict__ Wr,
                                const float* __restrict__ bias, float* __restrict__ out,
                                int M, int K, int N, int relu) {
    __shared__ float lds[4 * MAT_LDS];   // A1 | A2 | Wl | Wr chunks (36 KB)

    const int tid   = threadIdx.x;       // 0..127
    const int w     = tid >> 5;          // wave in block: 0..3
    const int lane  = tid & 31;
    const int l16   = lane & 15;
    const int half  = lane >> 4;         // 0: K={0,1}, rows 0-7 ; 1: K={2,3}, rows 8-15
    const int mtiles = M >> 4;
    const int msup  = blockIdx.y;        // 64-row super-tile
    const int nsup  = blockIdx.x;        // 64-col super-tile
    const int m0    = msup << 6;         // first (unclamped) row of super-tile
    const int mt0   = msup << 2;         // first 16-row tile index
    const int nbase = nsup << 6;         // first col of super-tile
    const int n0g   = nbase + (w << 4);  // this wave's 16-col tile

    // LDS fragment row offsets (dwords)
    int arow[4];
    #pragma unroll
    for (int t = 0; t < 4; ++t) arow[t] = ((t << 4) + l16) * LDS_STRIDE;
    const int wrow = ((w << 4) + l16) * LDS_STRIDE;

    const float* const la1 = lds;
    const float* const la2 = lds + MAT_LDS;
    const float* const lwl = lds + 2 * MAT_LDS;
    const float* const lwr = lds + 3 * MAT_LDS;

    v8f acc[4] = {};

    for (int kc = 0; kc < K; kc += KC) {
        __syncthreads();                 // protect LDS from previous chunk's readers
        // ---- async staging: 4 matrices x 64 rows x 8 float4 = 2048 b128 ----
        #pragma unroll
        for (int it = 0; it < 16; ++it) {
            const int m   = it >> 2;                       // matrix id (block-uniform)
            const int rem = ((it & 3) << 7) | tid;         // 0..511
            const int row = rem >> 3;                      // 0..63
            const int q   = rem & 7;                       // float4 within row
            const float* srcm;
            int grow;
            if (m == 0)      { grow = m0 + row; if (grow > M - 1) grow = M - 1; srcm = A1; }
            else if (m == 1) { grow = m0 + row; if (grow > M - 1) grow = M - 1; srcm = A2; }
            else if (m == 2) { grow = nbase + row; srcm = Wl; }
            else             { grow = nbase + row; srcm = Wr; }
            const float* gp = srcm + (size_t)grow * K + kc + (q << 2);
            unsigned lp = (unsigned)(uintptr_t)(lds + m * MAT_LDS
                                                + row * LDS_STRIDE + (q << 2));
            async_load_b128_to_lds(lp, gp);
        }
        wait_asynccnt0();                // own async writes to LDS complete
        __syncthreads();                 // all waves' writes visible

        // ---- compute: 8 K-steps, 8 WMMA each ----
        #pragma unroll
        for (int j = 0; j < KC / 4; ++j) {
            const int kk = (j << 2) + (half << 1);
            v2f bl = *(const v2f*)(lwl + wrow + kk);
            v2f br = *(const v2f*)(lwr + wrow + kk);
            #pragma unroll
            for (int t = 0; t < 4; ++t) {
                v2f a1 = *(const v2f*)(la1 + arow[t] + kk);
                acc[t] = wmma4_f32(a1, bl, acc[t]);
                v2f a2 = *(const v2f*)(la2 + arow[t] + kk);
                acc[t] = wmma4_f32(a2, br, acc[t]);
            }
        }
    }

    // ---- epilogue: bias (+ReLU), guarded wave-uniformly for tail tiles ----
    const float bv = bias[n0g + l16];
    #pragma unroll
    for (int t = 0; t < 4; ++t) {
        int mt = mt0 + t;
        if (mt < mtiles) {                         // wave-uniform guard
            int row0 = (mt << 4) + (half << 3);    // C/D layout: half -> rows 0-7 / 8-15
            #pragma unroll
            for (int r = 0; r < 8; ++r) {
                float v = acc[t][r] + bv;
                if (relu) v = fmaxf(v, 0.0f);
                out[(size_t)(row0 + r) * N + n0g + l16] = v;
            }
        }
    }
}

// ---------------------------------------------------------------------------
// Host launch
// ---------------------------------------------------------------------------
static inline int cdiv(int a, int b) { return (a + b - 1) / b; }

extern "C" void kernel_launch(void* const* d_in, const int* in_sizes, int n_in,
                              void* d_out, int out_size, void* d_ws, size_t ws_size,
                              hipStream_t stream) {
    const int IN = 128, HID = 256, OUTC = 128;
    const int N = in_sizes[0] / IN;     // 50000
    const int E = in_sizes[1] / 2;      // 640000

    const float* x   = (const float*)d_in[0];
    const int*   ei  = (const int*)d_in[1];
    const int*   esrc = ei;
    const int*   edst = ei + E;
    const float* W1l = (const float*)d_in[2];
    const float* b1  = (const float*)d_in[3];
    const float* W1r = (const float*)d_in[4];
    const float* W2l = (const float*)d_in[5];
    const float* b2  = (const float*)d_in[6];
    const float* W2r = (const float*)d_in[7];
    float* out = (float*)d_out;

    // Workspace: deg[N] | buf1[N*HID] (agg1/mean1, then agg2/mean2) | h[N*HID]
    float* deg  = (float*)d_ws;
    float* buf1 = deg + N;
    float* h    = buf1 + (size_t)N * HID;

    const int TB = 256;
    const int msupers = cdiv(N >> 4, 4);

    // 1) zero deg + agg1 (contiguous: N + N*IN floats)
    {
        int cnt = N * (1 + IN);
        zero_f32<<<cdiv(cnt, TB), TB, 0, stream>>>(deg, cnt);
    }
    // 2) degree count
    deg_count<<<cdiv(E, TB), TB, 0, stream>>>(edst, deg, E);
    // 3) scatter x -> agg1 (wave per edge)
    scatter_add<<<cdiv(E * 32, TB), TB, 0, stream>>>(x, esrc, edst, buf1, E, IN);
    // 4) deg -> 1/max(deg,1)
    rdeg_inv<<<cdiv(N, TB), TB, 0, stream>>>(deg, N);
    // 5) mean1 = agg1 * rdeg  (in place, logC = 7 for 128ch)
    scale_rows<<<cdiv(N * IN, TB), TB, 0, stream>>>(buf1, deg, N * IN, 7);
    // 6) h = relu(mean1 @ W1l^T + x @ W1r^T + b1)   [N, HID]
    {
        dim3 grid(HID / 64, msupers);
        sage_gemm_fused<<<grid, 128, 0, stream>>>(buf1, x, W1l, W1r, b1, h,
                                                  N, IN, HID, /*relu=*/1);
    }
    // 7) zero agg2 (reuse buf1, now N*HID floats)
    {
        int cnt = N * HID;
        zero_f32<<<cdiv(cnt, TB), TB, 0, stream>>>(buf1, cnt);
    }
    // 8) scatter h -> agg2
    scatter_add<<<cdiv(E * 32, TB), TB, 0, stream>>>(h, esrc, edst, buf1, E, HID);
    // 9) mean2 = agg2 * rdeg  (logC = 8 for 256ch)
    scale_rows<<<cdiv(N * HID, TB), TB, 0, stream>>>(buf1, deg, N * HID, 8);
    // 10) out = mean2 @ W2l^T + h @ W2r^T + b2   [N, OUTC]
    {
        dim3 grid(OUTC / 64, msupers);
        sage_gemm_fused<<<grid, 128, 0, stream>>>(buf1, h, W2l, W2r, b2, out,
                                                  N, HID, OUTC, /*relu=*/0);
    }
}